// DeformationGraph_23922967838893
// MI455X (gfx1250) — compile-verified
//
#include <hip/hip_runtime.h>
#include <math.h>

typedef float v2f __attribute__((ext_vector_type(2)));
typedef float v8f __attribute__((ext_vector_type(8)));

#define INV2SIG2 0.02f   // 1/(2*5*5)

// ---------------- Kernel 1: per-node prep (Rodrigues + affine terms) ----------------
// Z (flat 16 floats per node): [0..8]=R row-major, [9..11]=g=t+c-Rc, [12]=1, [13..15]=0
// C4[n] = (cx, cy, cz, |c|^2)
__global__ void dg_prep(const float* __restrict__ cps,
                        const float* __restrict__ rot,
                        const float* __restrict__ tr,
                        float4* __restrict__ Z4,
                        float4* __restrict__ C4,
                        int N)
{
    int n = blockIdx.x * blockDim.x + threadIdx.x;
    if (n >= N) return;
    float wx = rot[n*3+0], wy = rot[n*3+1], wz = rot[n*3+2];
    float th = sqrtf(wx*wx + wy*wy + wz*wz + 1e-12f);
    float it = 1.0f / th;
    float kx = wx*it, ky = wy*it, kz = wz*it;
    float s  = sinf(th);
    float c1 = 1.0f - cosf(th);
    float ksq = kx*kx + ky*ky + kz*kz;   // K@K = k k^T - (k.k) I  (exact identity)
    float R00 = 1.0f + c1*(kx*kx - ksq);
    float R01 = -s*kz + c1*(kx*ky);
    float R02 =  s*ky + c1*(kx*kz);
    float R10 =  s*kz + c1*(ky*kx);
    float R11 = 1.0f + c1*(ky*ky - ksq);
    float R12 = -s*kx + c1*(ky*kz);
    float R20 = -s*ky + c1*(kz*kx);
    float R21 =  s*kx + c1*(kz*ky);
    float R22 = 1.0f + c1*(kz*kz - ksq);
    float cx = cps[n*3+0], cy = cps[n*3+1], cz = cps[n*3+2];
    float tx = tr[n*3+0],  ty = tr[n*3+1],  tz = tr[n*3+2];
    float gx = tx + cx - (R00*cx + R01*cy + R02*cz);
    float gy = ty + cy - (R10*cx + R11*cy + R12*cz);
    float gz = tz + cz - (R20*cx + R21*cy + R22*cz);
    Z4[n*4+0] = make_float4(R00, R01, R02, R10);
    Z4[n*4+1] = make_float4(R11, R12, R20, R21);
    Z4[n*4+2] = make_float4(R22, gx, gy, gz);
    Z4[n*4+3] = make_float4(1.0f, 0.0f, 0.0f, 0.0f);
    C4[n]     = make_float4(cx, cy, cz, cx*cx + cy*cy + cz*cz);
}

// ---------------- Kernel 2: dual-WMMA warp kernel ----------------
// Per 16-node tile:
//   WMMA #1: Gram tile D[node,point] = c.p  (A=nodes Mx4, B=points 4x16, K=(x,y,z,0))
//   VALU   : w = exp(-(|p|^2+|c|^2-2 c.p)/2sig^2)  (8 per lane)
//   shfl   : rebuild W into A-fragment layout (lane = point row)
//   WMMA #2..5: OUT[point,col] += W(16x16) @ Z(16x13), K split 4x4, chained C accum
// Output D: lane = col, VGPR = point  -> 1KB/wave LDS transpose at the end.
__global__ __launch_bounds__(256)
void dg_warp(const float* __restrict__ points,
             const float4* __restrict__ Z4g,
             const float4* __restrict__ C4g,
             float* __restrict__ out,
             int N)
{
    __shared__ float4 zs[512*4];     // node table Z (32 KB), flat view: zsf[n*16+c]
    __shared__ float4 cs[512];       // (cx,cy,cz,|c|^2) (8 KB)
    __shared__ float4 ps[8*16];      // per-wave point tile: (x,y,z,|p|^2)
    __shared__ float  trans[8*256];  // per-wave D transpose buffer (8 KB)

    const int tid  = threadIdx.x;
    const int w    = tid >> 5;
    const int lane = tid & 31;
    const int l16  = lane & 15;
    const int h    = lane >> 4;          // 0 = K rows 0,1 ; 1 = K rows 2,3
    const int ptile = (blockIdx.x * 8 + w) * 16;

    __builtin_prefetch(points + (size_t)ptile * 3, 0, 0);  // global_prefetch_b8

    for (int i = tid; i < N*4; i += 256) zs[i] = Z4g[i];
    for (int i = tid; i < N;   i += 256) cs[i] = C4g[i];
    if (lane < 16) {
        int p = ptile + lane;
        float x = points[p*3+0], y = points[p*3+1], z = points[p*3+2];
        ps[w*16 + lane] = make_float4(x, y, z, x*x + y*y + z*z);
    }
    __syncthreads();

    const float* zsf = (const float*)zs;

    // B fragment for WMMA #1 (points): lane = column p = l16; K rows (x,y | z,0)
    float4 pv = ps[w*16 + l16];
    v2f bfrag;
    bfrag.x = h ? pv.z : pv.x;
    bfrag.y = h ? 0.0f : pv.y;
    const float pp = pv.w;

    v8f oacc = {};   // OUT accumulator: lane = col l16, VGPR r = point r+8h

    for (int nt = 0; nt < N; nt += 16) {
        // ---- WMMA #1: dot products ----
        float4 cv = cs[nt + l16];
        v2f afrag;
        afrag.x = h ? cv.z : cv.x;
        afrag.y = h ? 0.0f : cv.y;
        v8f zc = {};
        v8f dot = __builtin_amdgcn_wmma_f32_16x16x4_f32(
            false, afrag, false, bfrag, (short)0, zc, false, false);

        // ---- weights: lane holds W[node = nt+r+8h, point = l16] ----
        float wv[8];
#pragma unroll
        for (int r = 0; r < 8; ++r) {
            int n = nt + r + 8*h;
            float cc  = cs[n].w;
            float r2  = pp + cc - 2.0f * dot[r];
            wv[r] = __expf(-r2 * INV2SIG2);
        }
        float sh[8];
#pragma unroll
        for (int r = 0; r < 8; ++r) sh[r] = __shfl_xor(wv[r], 16, 32);

        // ---- A fragments (W, lane = point row) for the 4 K-chunks ----
        v2f a0, a1, a2, a3;
        a0.x = h ? sh[2] : wv[0];  a0.y = h ? sh[3] : wv[1];
        a1.x = h ? sh[6] : wv[4];  a1.y = h ? sh[7] : wv[5];
        a2.x = h ? wv[2] : sh[0];  a2.y = h ? wv[3] : sh[1];
        a3.x = h ? wv[6] : sh[4];  a3.y = h ? wv[7] : sh[5];

        // ---- B fragments (Z rows, lane = column c = l16) ----
        int base = (nt + 2*h) * 16 + l16;
        v2f b0, b1, b2, b3;
        b0.x = zsf[base +   0];  b0.y = zsf[base +  16];
        b1.x = zsf[base +  64];  b1.y = zsf[base +  80];
        b2.x = zsf[base + 128];  b2.y = zsf[base + 144];
        b3.x = zsf[base + 192];  b3.y = zsf[base + 208];

        // ---- WMMA #2..5: chained accumulation ----
        oacc = __builtin_amdgcn_wmma_f32_16x16x4_f32(false, a0, false, b0, (short)0, oacc, false, false);
        oacc = __builtin_amdgcn_wmma_f32_16x16x4_f32(false, a1, false, b1, (short)0, oacc, false, false);
        oacc = __builtin_amdgcn_wmma_f32_16x16x4_f32(false, a2, false, b2, (short)0, oacc, false, false);
        oacc = __builtin_amdgcn_wmma_f32_16x16x4_f32(false, a3, false, b3, (short)0, oacc, false, false);
    }

    // ---- transpose D through LDS: trans[point][col] ----
    float* tw = trans + w * 256;
#pragma unroll
    for (int r = 0; r < 8; ++r) tw[(r + 8*h) * 16 + l16] = oacc[r];
    __syncthreads();

    if (lane < 16) {
        const float* m = tw + l16 * 16;   // 0..8 = sum(w R), 9..11 = sum(w g), 12 = sum(w)
        float px = pv.x, py = pv.y, pz = pv.z;
        float inv = 1.0f / (m[12] + 1e-5f);
        int p = ptile + l16;
        out[p*3+0] = (m[0]*px + m[1]*py + m[2]*pz + m[9])  * inv;
        out[p*3+1] = (m[3]*px + m[4]*py + m[5]*pz + m[10]) * inv;
        out[p*3+2] = (m[6]*px + m[7]*py + m[8]*pz + m[11]) * inv;
    }
}

// ---------------- Kernel 3: edge regularizer (single block -> deterministic) ----------------
__global__ __launch_bounds__(256)
void dg_edges(const float* __restrict__ cps,
              const float* __restrict__ tr,
              const long long* __restrict__ edges,
              const float* __restrict__ Zf,      // rows of 16 floats; [0..8] = R row-major
              float* __restrict__ out_reg,
              int E)
{
    __shared__ float red[256];
    int tid = threadIdx.x;
    float sum = 0.0f;
    for (int e = tid; e < E; e += 256) {
        int i = (int)edges[2*e+0];
        int j = (int)edges[2*e+1];
        float cix = cps[i*3+0], ciy = cps[i*3+1], ciz = cps[i*3+2];
        float cjx = cps[j*3+0], cjy = cps[j*3+1], cjz = cps[j*3+2];
        float dx = cjx - cix, dy = cjy - ciy, dz = cjz - ciz;
        const float* R = Zf + (size_t)i * 16;
        float rx = R[0]*dx + R[1]*dy + R[2]*dz + cix + tr[i*3+0] - cjx - tr[j*3+0];
        float ry = R[3]*dx + R[4]*dy + R[5]*dz + ciy + tr[i*3+1] - cjy - tr[j*3+1];
        float rz = R[6]*dx + R[7]*dy + R[8]*dz + ciz + tr[i*3+2] - cjz - tr[j*3+2];
        sum += rx*rx + ry*ry + rz*rz;
    }
    red[tid] = sum;
    __syncthreads();
    for (int s = 128; s > 0; s >>= 1) {
        if (tid < s) red[tid] += red[tid + s];
        __syncthreads();
    }
    if (tid == 0) out_reg[0] = red[0];
}

extern "C" void kernel_launch(void* const* d_in, const int* in_sizes, int n_in,
                              void* d_out, int out_size, void* d_ws, size_t ws_size,
                              hipStream_t stream)
{
    const float* points     = (const float*)d_in[0];
    const float* cps        = (const float*)d_in[1];
    const float* rot        = (const float*)d_in[2];
    const float* tr         = (const float*)d_in[3];
    const long long* edges  = (const long long*)d_in[4];

    const int P = in_sizes[0] / 3;   // 65536
    const int N = in_sizes[1] / 3;   // 512
    const int E = in_sizes[4] / 2;   // 4096

    float* out = (float*)d_out;
    float4* Z4 = (float4*)d_ws;                                           // N*4 float4 (32 KB)
    float4* C4 = (float4*)((char*)d_ws + (size_t)N * 4 * sizeof(float4)); // N float4 (8 KB)

    dg_prep <<<(N + 255) / 256, 256, 0, stream>>>(cps, rot, tr, Z4, C4, N);
    dg_warp <<<P / 128, 256, 0, stream>>>(points, Z4, C4, out, N);
    dg_edges<<<1, 256, 0, stream>>>(cps, tr, edges, (const float*)d_ws,
                                    out + (size_t)P * 3, E);
}